// MPN_37091337568256
// MI455X (gfx1250) — compile-verified
//
#include <hip/hip_runtime.h>

#define NODES  50000
#define EDGES  800000
#define DIM    128
#define LAYERS 3

typedef __attribute__((ext_vector_type(2))) float v2f;
typedef __attribute__((ext_vector_type(8))) float v8f;

// ---------------------------------------------------------------------------
// Edge aggregation: agg[dst] += x[src].  One wave32 per edge, one float4 per
// lane (32 lanes x 4 = 128 features).  unsafeAtomicAdd -> global_atomic_add_f32.
// ---------------------------------------------------------------------------
__global__ __launch_bounds__(256) void scatter_add_kernel(
    const float* __restrict__ xin,
    const long long* __restrict__ ei,
    float* __restrict__ agg)
{
    const int edge = (blockIdx.x * blockDim.x + threadIdx.x) >> 5;
    const int lane = threadIdx.x & 31;
    if (edge >= EDGES) return;

    const long long src = ei[edge];
    const long long dst = ei[EDGES + edge];

    const float4 v = *(const float4*)(xin + src * DIM + lane * 4);
    float* ap = agg + dst * DIM + lane * 4;
    unsafeAtomicAdd(ap + 0, v.x);
    unsafeAtomicAdd(ap + 1, v.y);
    unsafeAtomicAdd(ap + 2, v.z);
    unsafeAtomicAdd(ap + 3, v.w);
}

// ---------------------------------------------------------------------------
// Fused GraphConv GEMM: out = agg @ Wrel + xin @ Wroot + bias, optional ReLU.
// Block = 256 threads = 8 waves.  Block b owns output rows [16b, 16b+16);
// wave w owns output columns [16w, 16w+16).  K reduction = 128 (agg) + 128 (x)
// in steps of 4 via V_WMMA_F32_16X16X4_F32 (fp32 matrix pipe, 64 WMMAs/wave).
//
// Fragment layouts per CDNA5 ISA 7.12.2 (fp32, wave32):
//   A 16x4 : lane half selects K pair: kb=(lane>>4)*2; row m=lane&15;
//            vgpr0=A[m][kb], vgpr1=A[m][kb+1]
//   B 4x16 : col n=lane&15; vgpr0=B[kb][n], vgpr1=B[kb+1][n]
//   C/D    : vgpr r -> row r (lanes 0-15) / row r+8 (lanes 16-31), col lane&15
// ---------------------------------------------------------------------------
template <bool RELU>
__global__ __launch_bounds__(256) void graphconv_gemm_kernel(
    const float* __restrict__ agg,
    const float* __restrict__ xin,
    const float* __restrict__ Wrel,
    const float* __restrict__ Wroot,
    const float* __restrict__ bias,
    float* __restrict__ out)
{
    const int lane = threadIdx.x & 31;
    const int wv   = threadIdx.x >> 5;       // 0..7 : column tile
    const int row0 = blockIdx.x * 16;
    const int col0 = wv * 16;
    const int m    = lane & 15;
    const int n    = lane & 15;
    const int kb   = (lane >> 4) * 2;        // 0 or 2

    const float* arow = agg + (size_t)(row0 + m) * DIM;
    const float* xrow = xin + (size_t)(row0 + m) * DIM;

    v8f acc = {};

    #pragma unroll 4
    for (int k = 0; k < DIM; k += 4) {
        v2f a, b;
        a.x = arow[k + kb];
        a.y = arow[k + kb + 1];
        b.x = Wrel[(size_t)(k + kb)     * DIM + col0 + n];
        b.y = Wrel[(size_t)(k + kb + 1) * DIM + col0 + n];
        acc = __builtin_amdgcn_wmma_f32_16x16x4_f32(
            false, a, false, b, (short)0, acc, false, false);
    }
    #pragma unroll 4
    for (int k = 0; k < DIM; k += 4) {
        v2f a, b;
        a.x = xrow[k + kb];
        a.y = xrow[k + kb + 1];
        b.x = Wroot[(size_t)(k + kb)     * DIM + col0 + n];
        b.y = Wroot[(size_t)(k + kb + 1) * DIM + col0 + n];
        acc = __builtin_amdgcn_wmma_f32_16x16x4_f32(
            false, a, false, b, (short)0, acc, false, false);
    }

    const float bb   = bias[col0 + n];
    const int   moff = (lane >> 4) * 8;      // C/D layout: lanes 16-31 -> rows +8
    #pragma unroll
    for (int r = 0; r < 8; ++r) {
        float v = acc[r] + bb;
        if (RELU) v = v > 0.0f ? v : 0.0f;
        out[(size_t)(row0 + moff + r) * DIM + col0 + n] = v;
    }
}

// ---------------------------------------------------------------------------
// d_in: [0]=x (50000x128 f32), [1]=edge_index (2x800000 i64),
//       [2]=W_rel (3x128x128), [3]=W_root (3x128x128), [4]=b (3x128)
// d_out: 50000x128 f32
// d_ws : agg (25.6MB) | bufA (25.6MB) | bufB (25.6MB)
// ---------------------------------------------------------------------------
extern "C" void kernel_launch(void* const* d_in, const int* in_sizes, int n_in,
                              void* d_out, int out_size, void* d_ws, size_t ws_size,
                              hipStream_t stream) {
    (void)in_sizes; (void)n_in; (void)out_size; (void)ws_size;

    const float*     x      = (const float*)d_in[0];
    const long long* ei     = (const long long*)d_in[1];
    const float*     W_rel  = (const float*)d_in[2];
    const float*     W_root = (const float*)d_in[3];
    const float*     b      = (const float*)d_in[4];
    float*           out    = (float*)d_out;

    const size_t nd = (size_t)NODES * DIM;
    float* agg  = (float*)d_ws;
    float* bufA = agg + nd;
    float* bufB = bufA + nd;

    const float* in = x;
    for (int l = 0; l < LAYERS; ++l) {
        hipMemsetAsync(agg, 0, nd * sizeof(float), stream);

        scatter_add_kernel<<<EDGES / 8, 256, 0, stream>>>(in, ei, agg);

        float* o = (l == 0) ? bufA : (l == 1) ? bufB : out;
        if (l < LAYERS - 1) {
            graphconv_gemm_kernel<true><<<NODES / 16, 256, 0, stream>>>(
                agg, in, W_rel + (size_t)l * DIM * DIM,
                W_root + (size_t)l * DIM * DIM, b + (size_t)l * DIM, o);
        } else {
            graphconv_gemm_kernel<false><<<NODES / 16, 256, 0, stream>>>(
                agg, in, W_rel + (size_t)l * DIM * DIM,
                W_root + (size_t)l * DIM * DIM, b + (size_t)l * DIM, o);
        }
        in = o;
    }
}